// AGCRN_7619271983544
// MI455X (gfx1250) — compile-verified
//
#include <hip/hip_runtime.h>
#include <hip/hip_bf16.h>
#include <cstddef>

// ---------------- problem constants ----------------
#define NN_ 2048      // nodes
#define BB_ 16        // batch
#define CIN_ 2
#define HID_ 64
#define DD_ 16        // embedding dim
#define CI_ (CIN_ + HID_)     // 66
#define COLS_ (BB_ * CI_)     // 1056 (batch folded into GEMM columns)
#define G2_ (2 * HID_)        // 128
#define KI_ (2 * CI_)         // 132 (K=2 chebyshev * 66 channels)

typedef float v2f __attribute__((ext_vector_type(2)));
typedef float v8f __attribute__((ext_vector_type(8)));

// ============================================================
// Kernel 1: S = relu(E @ E^T) via f32 WMMA. 128x128 tiles of 16x16,
// one tile per wave, K=16 -> 4 wmma per tile.
// ============================================================
__global__ void k_ee_wmma(const float* __restrict__ E, float* __restrict__ S) {
    const int wavesPerBlk = blockDim.x >> 5;
    const int wave = blockIdx.x * wavesPerBlk + (threadIdx.x >> 5);
    const int lane = threadIdx.x & 31;
    const int lo = lane & 15, hi = lane >> 4;
    const int tilesN = NN_ / 16;
    const int tm = (wave / tilesN) * 16;
    const int tn = (wave % tilesN) * 16;

    v8f c = {};
#pragma unroll
    for (int kk = 0; kk < DD_; kk += 4) {
        // A: 16x4 f32, lane(0-15) row=lo K={kk,kk+1}; lane(16-31) K={kk+2,kk+3}
        v2f a = *(const v2f*)(E + (size_t)(tm + lo) * DD_ + kk + 2 * hi);
        // B = E^T: B[k][n] = E[n][k] -> same contiguous-pair pattern
        v2f b = *(const v2f*)(E + (size_t)(tn + lo) * DD_ + kk + 2 * hi);
        c = __builtin_amdgcn_wmma_f32_16x16x4_f32(false, a, false, b,
                                                  (short)0, c, false, false);
    }
#pragma unroll
    for (int j = 0; j < 8; ++j) {
        float v = c[j];
        v = v > 0.f ? v : 0.f;                        // fused relu
        S[(size_t)(tm + j + 8 * hi) * NN_ + (tn + lo)] = v;
    }
}

// ============================================================
// Kernel 2: row-wise softmax of S (2048 rows), one block per row.
// ============================================================
__global__ void k_softmax(float* __restrict__ S) {
    const int row = blockIdx.x;
    float* p = S + (size_t)row * NN_;
    __shared__ float red[256];
    const int t = threadIdx.x;

    float m = -1e30f;
    for (int i = t; i < NN_; i += 256) m = fmaxf(m, p[i]);
    red[t] = m; __syncthreads();
    for (int s = 128; s > 0; s >>= 1) {
        if (t < s) red[t] = fmaxf(red[t], red[t + s]);
        __syncthreads();
    }
    m = red[0]; __syncthreads();

    float sum = 0.f;
    for (int i = t; i < NN_; i += 256) { float e = __expf(p[i] - m); p[i] = e; sum += e; }
    red[t] = sum; __syncthreads();
    for (int s = 128; s > 0; s >>= 1) {
        if (t < s) red[t] += red[t + s];
        __syncthreads();
    }
    const float inv = 1.f / red[0];
    for (int i = t; i < NN_; i += 256) p[i] *= inv;
}

// ============================================================
// Kernel 3a: pack XcatT[col][n], col = b*66+c : concat(X, state), transposed
// so the big GEMM's B operand is read along K contiguously.
// ============================================================
__global__ void k_pack_gate(const float* __restrict__ X, const float* __restrict__ state,
                            float* __restrict__ XT) {
    const size_t idx = (size_t)blockIdx.x * 256 + threadIdx.x;
    if (idx >= (size_t)COLS_ * NN_) return;
    const int n = (int)(idx % NN_);
    const int col = (int)(idx / NN_);
    const int b = col / CI_, c = col % CI_;
    float v = (c < CIN_) ? X[((size_t)b * NN_ + n) * CIN_ + c]
                         : state[((size_t)b * NN_ + n) * HID_ + (c - CIN_)];
    XT[idx] = v;
}

// Kernel 3b: pack CandT[col][n] = concat(X, z*state), transposed.
__global__ void k_pack_cand(const float* __restrict__ X, const float* __restrict__ state,
                            const float* __restrict__ Z, float* __restrict__ XT) {
    const size_t idx = (size_t)blockIdx.x * 256 + threadIdx.x;
    if (idx >= (size_t)COLS_ * NN_) return;
    const int n = (int)(idx % NN_);
    const int col = (int)(idx / NN_);
    const int b = col / CI_, c = col % CI_;
    float v;
    if (c < CIN_) v = X[((size_t)b * NN_ + n) * CIN_ + c];
    else {
        const size_t si = ((size_t)b * NN_ + n) * HID_ + (c - CIN_);
        v = Z[si] * state[si];
    }
    XT[idx] = v;
}

// ============================================================
// Kernel 4: Y[n][col] = sum_m S[n][m] * XT[col][m]
// M=2048, K=2048, Ncols=1056. Register-blocked: each wave computes a
// 64x32 macro-tile = 4x2 WMMA tiles. Per K-step of 4: 6 b64 loads feed
// 8 v_wmma_f32_16x16x4_f32 (192 B per wmma instead of 512 B).
// Per-lane global_prefetch 2KB ahead on each of the 6 row streams.
// ============================================================
#define MT_ 4   // 16x16 tiles in M per wave
#define NT_ 2   // 16x16 tiles in N per wave
#define PF_ 512 // prefetch distance in floats (2KB)

__global__ void k_gemm_sx(const float* __restrict__ S, const float* __restrict__ XT,
                          float* __restrict__ Y) {
    const int wavesPerBlk = blockDim.x >> 5;
    const int wave = blockIdx.x * wavesPerBlk + (threadIdx.x >> 5);
    const int lane = threadIdx.x & 31;
    const int lo = lane & 15, hi = lane >> 4;
    const int colJobs = COLS_ / (16 * NT_);        // 33
    const int tm = (wave / colJobs) * (16 * MT_);  // 0..2047 step 64
    const int tc = (wave % colJobs) * (16 * NT_);  // 0..1055 step 32

    const float* aptr[MT_];
#pragma unroll
    for (int i = 0; i < MT_; ++i)
        aptr[i] = S + (size_t)(tm + 16 * i + lo) * NN_ + 2 * hi;
    const float* bptr[NT_];
#pragma unroll
    for (int j = 0; j < NT_; ++j)
        bptr[j] = XT + (size_t)(tc + 16 * j + lo) * NN_ + 2 * hi;

    v8f acc[MT_][NT_] = {};
#pragma unroll 8
    for (int kk = 0; kk < NN_; kk += 4) {
        if ((kk & 31) == 0) {      // once per 128B of per-lane stream advance
            const int pf = kk + PF_;
            if (pf < NN_) {
#pragma unroll
                for (int i = 0; i < MT_; ++i) __builtin_prefetch(aptr[i] + pf, 0, 3);
#pragma unroll
                for (int j = 0; j < NT_; ++j) __builtin_prefetch(bptr[j] + pf, 0, 3);
            }
        }
        v2f a[MT_], b[NT_];
#pragma unroll
        for (int i = 0; i < MT_; ++i) a[i] = *(const v2f*)(aptr[i] + kk);
#pragma unroll
        for (int j = 0; j < NT_; ++j) b[j] = *(const v2f*)(bptr[j] + kk);
#pragma unroll
        for (int i = 0; i < MT_; ++i)
#pragma unroll
            for (int j = 0; j < NT_; ++j)
                acc[i][j] = __builtin_amdgcn_wmma_f32_16x16x4_f32(
                    false, a[i], false, b[j], (short)0, acc[i][j], false, false);
    }

#pragma unroll
    for (int i = 0; i < MT_; ++i)
#pragma unroll
        for (int j = 0; j < NT_; ++j)
#pragma unroll
            for (int jj = 0; jj < 8; ++jj)
                Y[(size_t)(tm + 16 * i + jj + 8 * hi) * COLS_ + (tc + 16 * j + lo)]
                    = acc[i][j][jj];
}

// ============================================================
// Kernel 5: per-node gate. Block = (node n, half) ; 128 threads = 4 waves.
// Builds W_n[132][64] = sum_d E[n,d]*gate_W[d,:,half*64+:64] in LDS,
// A_n[16][132] (batch x flattened (k,i)) in LDS, then 4 WMMA col-tiles.
// Epilogue: sigmoid(acc + bias) -> z (half 0) / r (half 1).
// ============================================================
__global__ void k_node_gate(const float* __restrict__ XT, const float* __restrict__ Y1,
                            const float* __restrict__ E,  const float* __restrict__ gW,
                            const float* __restrict__ gb,
                            float* __restrict__ Z, float* __restrict__ R) {
    const int n = blockIdx.x;
    const int half = blockIdx.y;               // 0 -> z cols, 1 -> r cols
    __shared__ float e[DD_];
    __shared__ float Wn[KI_][HID_];            // 132 x 64
    __shared__ float Ag[BB_][KI_];             // 16 x 132
    __shared__ float bias[HID_];
    const int t = threadIdx.x;

    if (t < DD_) e[t] = E[(size_t)n * DD_ + t];
    __syncthreads();

    for (int idx = t; idx < KI_ * HID_; idx += 128) {
        const int ki = idx / HID_, oc = idx % HID_;
        const float* wp = gW + (size_t)ki * G2_ + half * HID_ + oc;
        float acc = 0.f;
#pragma unroll
        for (int d = 0; d < DD_; ++d) acc += e[d] * wp[(size_t)d * KI_ * G2_];
        Wn[ki][oc] = acc;
    }
    if (t < HID_) {
        float acc = 0.f;
#pragma unroll
        for (int d = 0; d < DD_; ++d) acc += e[d] * gb[(size_t)d * G2_ + half * HID_ + t];
        bias[t] = acc;
    }
    for (int idx = t; idx < BB_ * KI_; idx += 128) {
        const int b = idx / KI_, ki = idx % KI_;
        float v = (ki < CI_) ? XT[(size_t)(b * CI_ + ki) * NN_ + n]           // k=0: identity term
                             : Y1[(size_t)n * COLS_ + b * CI_ + (ki - CI_)];  // k=1: supports term
        Ag[b][ki] = v;
    }
    __syncthreads();

    const int w = t >> 5, lane = t & 31, lo = lane & 15, hi = lane >> 4;
    const int to = w * 16;
    v8f c = {};
#pragma unroll 4
    for (int kk = 0; kk < KI_; kk += 4) {
        v2f a; a.x = Ag[lo][kk + 2 * hi];       a.y = Ag[lo][kk + 2 * hi + 1];
        v2f b; b.x = Wn[kk + 2 * hi][to + lo];  b.y = Wn[kk + 2 * hi + 1][to + lo];
        c = __builtin_amdgcn_wmma_f32_16x16x4_f32(false, a, false, b,
                                                  (short)0, c, false, false);
    }
#pragma unroll
    for (int j = 0; j < 8; ++j) {
        const int bb = j + 8 * hi;             // batch index
        const int oc = to + lo;                // col within this half
        float v = c[j] + bias[oc];
        v = 1.f / (1.f + __expf(-v));          // sigmoid
        float* dst = (half == 0) ? Z : R;
        dst[((size_t)bb * NN_ + n) * HID_ + oc] = v;
    }
}

// ============================================================
// Kernel 6: per-node update + GRU combine. Block = node n; 128 threads.
// hc = tanh(acc + bias);  h = r*state + (1-r)*hc  -> d_out.
// ============================================================
__global__ void k_node_upd(const float* __restrict__ CT, const float* __restrict__ Y2,
                           const float* __restrict__ E,  const float* __restrict__ uW,
                           const float* __restrict__ ub,
                           const float* __restrict__ R,  const float* __restrict__ state,
                           float* __restrict__ H) {
    const int n = blockIdx.x;
    __shared__ float e[DD_];
    __shared__ float Wn[KI_][HID_];            // 132 x 64
    __shared__ float Ag[BB_][KI_];             // 16 x 132
    __shared__ float bias[HID_];
    const int t = threadIdx.x;

    if (t < DD_) e[t] = E[(size_t)n * DD_ + t];
    __syncthreads();

    for (int idx = t; idx < KI_ * HID_; idx += 128) {
        const int ki = idx / HID_, oc = idx % HID_;
        const float* wp = uW + (size_t)ki * HID_ + oc;
        float acc = 0.f;
#pragma unroll
        for (int d = 0; d < DD_; ++d) acc += e[d] * wp[(size_t)d * KI_ * HID_];
        Wn[ki][oc] = acc;
    }
    if (t < HID_) {
        float acc = 0.f;
#pragma unroll
        for (int d = 0; d < DD_; ++d) acc += e[d] * ub[(size_t)d * HID_ + t];
        bias[t] = acc;
    }
    for (int idx = t; idx < BB_ * KI_; idx += 128) {
        const int b = idx / KI_, ki = idx % KI_;
        float v = (ki < CI_) ? CT[(size_t)(b * CI_ + ki) * NN_ + n]
                             : Y2[(size_t)n * COLS_ + b * CI_ + (ki - CI_)];
        Ag[b][ki] = v;
    }
    __syncthreads();

    const int w = t >> 5, lane = t & 31, lo = lane & 15, hi = lane >> 4;
    const int to = w * 16;
    v8f c = {};
#pragma unroll 4
    for (int kk = 0; kk < KI_; kk += 4) {
        v2f a; a.x = Ag[lo][kk + 2 * hi];       a.y = Ag[lo][kk + 2 * hi + 1];
        v2f b; b.x = Wn[kk + 2 * hi][to + lo];  b.y = Wn[kk + 2 * hi + 1][to + lo];
        c = __builtin_amdgcn_wmma_f32_16x16x4_f32(false, a, false, b,
                                                  (short)0, c, false, false);
    }
#pragma unroll
    for (int j = 0; j < 8; ++j) {
        const int bb = j + 8 * hi;
        const int oc = to + lo;
        const size_t si = ((size_t)bb * NN_ + n) * HID_ + oc;
        const float hc = tanhf(c[j] + bias[oc]);
        const float r = R[si];
        H[si] = r * state[si] + (1.f - r) * hc;
    }
}

// ============================================================
// Host side
// ============================================================
extern "C" void kernel_launch(void* const* d_in, const int* in_sizes, int n_in,
                              void* d_out, int out_size, void* d_ws, size_t ws_size,
                              hipStream_t stream) {
    const float* X     = (const float*)d_in[0];   // (B,N,CIN)
    const float* state = (const float*)d_in[1];   // (B,N,HID)
    const float* E     = (const float*)d_in[2];   // (N,D)
    const float* gW    = (const float*)d_in[3];   // (D,K,66,128)
    const float* gb    = (const float*)d_in[4];   // (D,128)
    const float* uW    = (const float*)d_in[5];   // (D,K,66,64)
    const float* ub    = (const float*)d_in[6];   // (D,64)
    float* H = (float*)d_out;                     // (B,N,HID)

    float* ws = (float*)d_ws;
    float* S  = ws;                                        // 2048*2048
    float* XT = S  + (size_t)NN_ * NN_;                    // 1056*2048 (Xcat^T, reused for Cand^T)
    float* Y  = XT + (size_t)COLS_ * NN_;                  // 2048*1056 (Y1, reused for Y2)
    float* Z  = Y  + (size_t)NN_ * COLS_;                  // 16*2048*64
    float* R  = Z  + (size_t)BB_ * NN_ * HID_;             // 16*2048*64
    // total ~51 MB of f32 scratch

    const int eeTiles   = (NN_ / 16) * (NN_ / 16);         // 16384 tiles, 8 waves/block
    const int gemmJobs  = (NN_ / (16 * MT_)) * (COLS_ / (16 * NT_)); // 32*33 = 1056 waves
    const int packBlks  = (int)(((size_t)COLS_ * NN_ + 255) / 256);

    // 1) S = relu(E E^T)
    k_ee_wmma<<<eeTiles / 8, 256, 0, stream>>>(E, S);
    // 2) row softmax
    k_softmax<<<NN_, 256, 0, stream>>>(S);
    // 3) Xcat^T
    k_pack_gate<<<packBlks, 256, 0, stream>>>(X, state, XT);
    // 4) Y1 = S @ Xcat
    k_gemm_sx<<<gemmJobs / 8, 256, 0, stream>>>(S, XT, Y);
    // 5) z, r = sigmoid(per-node gate GEMM + bias)
    k_node_gate<<<dim3(NN_, 2), 128, 0, stream>>>(XT, Y, E, gW, gb, Z, R);
    // 6) Cand^T (overwrites XT)
    k_pack_cand<<<packBlks, 256, 0, stream>>>(X, state, Z, XT);
    // 7) Y2 = S @ Cand (overwrites Y)
    k_gemm_sx<<<gemmJobs / 8, 256, 0, stream>>>(S, XT, Y);
    // 8) hc = tanh(...); h = r*state + (1-r)*hc
    k_node_upd<<<NN_, 128, 0, stream>>>(XT, Y, E, uW, ub, R, state, H);
}